// NNConv_59528246723314
// MI455X (gfx1250) — compile-verified
//
#include <hip/hip_runtime.h>
#include <hip/hip_bf16.h>

// ---------------------------------------------------------------------------
// NNConv (edge-conditioned graph conv), MI455X / gfx1250, wave32, fp32 WMMA.
//
// Re-associated algorithm (8x less FLOPs than the naive fused form):
//   M[n,k,o]   = sum_i feat[n,i] * W_edge[k, i*32+o]      (WMMA GEMM, 1.07 GFLOP)
//   Bterm[n,o] = sum_i feat[n,i] * b_edge[i*32+o]         (WMMA)
//   out[n,o]   = sum_i feat[n,i] * root[i,o] + bias[o]    (WMMA, init)
//   msg[e,o]   = sum_k efeat[e,k] * M[src[e],k,o] + Bterm[src[e],o]
//   out[dst[e],o] += msg[e,o]                              (f32 global atomics)
// M is 64 MB -> resident in the 192 MB L2, so per-edge gathers stay on-chip.
// ---------------------------------------------------------------------------

#define NN  16384   // nodes
#define EE  131072  // edges
#define INF 32
#define OUTF 32

typedef __attribute__((ext_vector_type(2))) float v2f;
typedef __attribute__((ext_vector_type(8))) float v8f;

// ---------------------------------------------------------------------------
// Kernel 1: M[n, k*32+o] = sum_i feat[n,i] * W_edge[k, i*32+o]
// GEMM [16384 x 32] @ [32 x 1024]; B[i, c] = W_edge[c>>5, i*32 + (c&31)].
// Grid: 8192 blocks x 256 thr (8 waves). Wave -> one 16x16 C tile, K=32 -> 8 WMMA.
// ---------------------------------------------------------------------------
__global__ void __launch_bounds__(256)
node_transform_wmma(const float* __restrict__ feat,
                    const float* __restrict__ W_edge,
                    float* __restrict__ Mbuf) {
    const int nt   = blockIdx.x >> 3;          // node tile 0..1023 (16 nodes)
    const int cb   = blockIdx.x & 7;           // column block
    const int wave = threadIdx.x >> 5;         // 0..7
    const int ct   = cb * 8 + wave;            // column tile 0..63 (16 cols)
    const int lane = threadIdx.x & 31;
    const int n    = lane & 15;                // B/C column, A row (m)
    const int g    = lane >> 4;                // lane group

    const int k_e = ct >> 1;                   // 16-wide c-tile lies in one k row
    const int ob  = (ct & 1) << 4;             // o base within W_edge row

    v8f c = {};
#pragma unroll
    for (int s = 0; s < 8; ++s) {              // K steps of 4
        v2f a, b;
#pragma unroll
        for (int r = 0; r < 2; ++r) {
            const int kk = s * 4 + 2 * g + r;  // K index (row of A-col / B-row)
            a[r] = feat[(nt * 16 + n) * INF + kk];
            b[r] = W_edge[k_e * 1024 + kk * 32 + ob + n];
        }
        c = __builtin_amdgcn_wmma_f32_16x16x4_f32(false, a, false, b,
                                                  (short)0, c, false, false);
    }
#pragma unroll
    for (int r = 0; r < 8; ++r)                // C row m = 8*g + r, col = n
        Mbuf[(nt * 16 + 8 * g + r) * 1024 + ct * 16 + n] = c[r];
}

// ---------------------------------------------------------------------------
// Kernel 2: Bterm = feat @ b_edge'   and   out = feat @ root + bias (init).
// Grid: 1024 blocks x 128 thr (4 waves): wave>>1 selects target, wave&1 col half.
// ---------------------------------------------------------------------------
__global__ void __launch_bounds__(128)
node_small_wmma(const float* __restrict__ feat,
                const float* __restrict__ b_edge,
                const float* __restrict__ root,
                const float* __restrict__ bias,
                float* __restrict__ Bterm,
                float* __restrict__ out) {
    const int nt     = blockIdx.x;
    const int wave   = threadIdx.x >> 5;
    const int lane   = threadIdx.x & 31;
    const int target = wave >> 1;              // 0: Bterm(b_edge) 1: out(root)
    const int oh     = (wave & 1) << 4;        // column half offset
    const int n      = lane & 15;
    const int g      = lane >> 4;

    const float* Bm = (target == 0) ? b_edge : root;   // both [32 x 32] row-major

    v8f c = {};
#pragma unroll
    for (int s = 0; s < 8; ++s) {
        v2f a, b;
#pragma unroll
        for (int r = 0; r < 2; ++r) {
            const int kk = s * 4 + 2 * g + r;
            a[r] = feat[(nt * 16 + n) * INF + kk];
            b[r] = Bm[kk * 32 + oh + n];
        }
        c = __builtin_amdgcn_wmma_f32_16x16x4_f32(false, a, false, b,
                                                  (short)0, c, false, false);
    }
    if (target == 0) {
#pragma unroll
        for (int r = 0; r < 8; ++r)
            Bterm[(nt * 16 + 8 * g + r) * OUTF + oh + n] = c[r];
    } else {
        const float bv = bias[oh + n];
#pragma unroll
        for (int r = 0; r < 8; ++r)
            out[(nt * 16 + 8 * g + r) * OUTF + oh + n] = c[r] + bv;
    }
}

// ---------------------------------------------------------------------------
// Kernel 3: one wave32 per edge; lane = output feature o.
// msg[e,o] = Bterm[src,o] + sum_k efeat[e,k] * M[src, k*32+o]; atomic scatter.
// M rows (4 KB, contiguous) are L2-resident -> on-chip gather bandwidth.
// ---------------------------------------------------------------------------
__global__ void __launch_bounds__(256)
edge_gather(const float* __restrict__ efeat,
            const int*   __restrict__ src,
            const int*   __restrict__ dst,
            const float* __restrict__ Mbuf,
            const float* __restrict__ Bterm,
            float*       __restrict__ out) {
    const int e = blockIdx.x * 8 + (threadIdx.x >> 5);
    const int o = threadIdx.x & 31;

    const int sN = src[e];
    const int dN = dst[e];

    const float ef = efeat[e * 32 + o];        // one coalesced load, shuffled below
    const float* __restrict__ Mrow = Mbuf + (size_t)sN * 1024;

    __builtin_prefetch(Mrow, 0, 1);            // global_prefetch_b8
    __builtin_prefetch(Mrow + 512, 0, 1);

    float acc = Bterm[sN * 32 + o];
#pragma unroll
    for (int k = 0; k < 32; ++k) {
        const float efk = __shfl(ef, k, 32);   // wave32 broadcast (ds_bpermute)
        acc = fmaf(efk, Mrow[k * 32 + o], acc);
    }
    atomicAdd(out + (size_t)dN * 32 + o, acc); // global_atomic_add_f32
}

// ---------------------------------------------------------------------------
// Fallback path (only if workspace < 66 MB): correct but heavier fused VALU.
// ---------------------------------------------------------------------------
__global__ void __launch_bounds__(256)
out_init_fallback(const float* __restrict__ feat,
                  const float* __restrict__ root,
                  const float* __restrict__ bias,
                  float* __restrict__ out) {
    const int nN = blockIdx.x * 8 + (threadIdx.x >> 5);
    const int o  = threadIdx.x & 31;
    const float f = feat[nN * 32 + o];
    float acc = bias[o];
#pragma unroll
    for (int i = 0; i < 32; ++i)
        acc = fmaf(__shfl(f, i, 32), root[i * 32 + o], acc);
    out[nN * 32 + o] = acc;
}

__global__ void __launch_bounds__(256)
edge_fused_fallback(const float* __restrict__ feat,
                    const float* __restrict__ efeat,
                    const float* __restrict__ W_edge,
                    const float* __restrict__ b_edge,
                    const int*   __restrict__ src,
                    const int*   __restrict__ dst,
                    float*       __restrict__ out) {
    const int e = blockIdx.x * 8 + (threadIdx.x >> 5);
    const int o = threadIdx.x & 31;
    const int sN = src[e];
    const int dN = dst[e];
    const float f  = feat[sN * 32 + o];
    const float ef = efeat[e * 32 + o];
    float acc = 0.0f;
    for (int i = 0; i < 32; ++i) {
        float w = b_edge[i * 32 + o];
#pragma unroll 8
        for (int k = 0; k < 32; ++k)
            w = fmaf(__shfl(ef, k, 32), W_edge[k * 1024 + i * 32 + o], w);
        acc = fmaf(__shfl(f, i, 32), w, acc);
    }
    atomicAdd(out + (size_t)dN * 32 + o, acc);
}

// ---------------------------------------------------------------------------
// Host launcher. Inputs (setup_inputs order):
//   0 feat[N,32] f32   1 efeat[E,32] f32   2 W_edge[32,1024] f32
//   3 b_edge[1024] f32 4 root[32,32] f32   5 bias[32] f32
//   6 src[E] int       7 dst[E] int        out: [N,32] f32
// ---------------------------------------------------------------------------
extern "C" void kernel_launch(void* const* d_in, const int* in_sizes, int n_in,
                              void* d_out, int out_size, void* d_ws, size_t ws_size,
                              hipStream_t stream) {
    const float* feat   = (const float*)d_in[0];
    const float* efeat  = (const float*)d_in[1];
    const float* W_edge = (const float*)d_in[2];
    const float* b_edge = (const float*)d_in[3];
    const float* root   = (const float*)d_in[4];
    const float* bias   = (const float*)d_in[5];
    const int*   src    = (const int*)d_in[6];
    const int*   dst    = (const int*)d_in[7];
    float*       out    = (float*)d_out;

    const size_t needM = (size_t)NN * 1024 * sizeof(float);   // 64 MB
    const size_t needB = (size_t)NN * OUTF * sizeof(float);   //  2 MB

    if (ws_size >= needM + needB) {
        float* Mbuf  = (float*)d_ws;
        float* Bterm = Mbuf + (size_t)NN * 1024;

        node_transform_wmma<<<(NN / 16) * 8, 256, 0, stream>>>(feat, W_edge, Mbuf);
        node_small_wmma<<<NN / 16, 128, 0, stream>>>(feat, b_edge, root, bias,
                                                     Bterm, out);
        edge_gather<<<EE / 8, 256, 0, stream>>>(efeat, src, dst, Mbuf, Bterm, out);
    } else {
        out_init_fallback<<<NN / 8, 256, 0, stream>>>(feat, root, bias, out);
        edge_fused_fallback<<<EE / 8, 256, 0, stream>>>(feat, efeat, W_edge, b_edge,
                                                        src, dst, out);
    }
}